// MultiViewHyperConvNetwork_8864812499581
// MI455X (gfx1250) — compile-verified
//
#include <hip/hip_runtime.h>
#include <hip/hip_bf16.h>
#include <stdint.h>

// Problem constants (match reference)
#define P_N        200000
#define U_N        50000
#define D_N        64
#define NNZ_N      2000000
#define NUM_LAYERS 3

// ---------------------------------------------------------------------------
// Threefry-2x32 block cipher (bit-exact with jax._src.prng.threefry2x32)
// ---------------------------------------------------------------------------
__host__ __device__ inline unsigned tf_rotl(unsigned x, int r) {
  return (x << r) | (x >> (32 - r));
}

__host__ __device__ inline void threefry2x32_block(unsigned k0, unsigned k1,
                                                   unsigned x0, unsigned x1,
                                                   unsigned& y0, unsigned& y1) {
  const unsigned ks2 = k0 ^ k1 ^ 0x1BD11BDAu;
  x0 += k0; x1 += k1;
#define TF_R(r) { x0 += x1; x1 = tf_rotl(x1, (r)); x1 ^= x0; }
  TF_R(13) TF_R(15) TF_R(26) TF_R(6)
  x0 += k1;  x1 += ks2 + 1u;
  TF_R(17) TF_R(29) TF_R(16) TF_R(24)
  x0 += ks2; x1 += k0 + 2u;
  TF_R(13) TF_R(15) TF_R(26) TF_R(6)
  x0 += k0;  x1 += k1 + 3u;
  TF_R(17) TF_R(29) TF_R(16) TF_R(24)
  x0 += k1;  x1 += ks2 + 4u;
  TF_R(13) TF_R(15) TF_R(26) TF_R(6)
  x0 += ks2; x1 += k0 + 5u;
#undef TF_R
  y0 = x0; y1 = x1;
}

// ---------------------------------------------------------------------------
// CDNA5 helpers (gfx1250): async-to-LDS (ASYNCcnt), TDM (TENSORcnt),
// native device-scope f32 atomic add.
// ---------------------------------------------------------------------------
// Low 32 bits of a flat LDS address are the workgroup-relative LDS byte
// offset (ISA 10.2 aperture mapping) == what async/TDM LDS-address fields use.
__device__ inline unsigned lds_addr32(const void* p) {
  return (unsigned)(uintptr_t)p;
}

__device__ inline void async_load_b32_to_lds(unsigned ldsOff, const void* gaddr) {
  asm volatile("global_load_async_to_lds_b32 %0, %1, off"
               :
               : "v"(ldsOff), "v"((unsigned long long)(uintptr_t)gaddr)
               : "memory");
}

__device__ inline void wait_asynccnt0() {
  asm volatile("s_wait_asynccnt 0" ::: "memory");
}

// Non-returning native f32 atomic add, device scope (RMW at L2 atomic units).
__device__ inline void atomic_add_f32_dev(float* p, float v) {
  asm volatile("global_atomic_add_f32 %0, %1, off scope:SCOPE_DEV"
               :
               : "v"((unsigned long long)(uintptr_t)p), "v"(v)
               : "memory");
}

// TDM: 1-D tile load Global->LDS via tensor descriptor (ISA ch.8 D# layout).
// tensorLen = elements remaining from gptr (OOB reads return zero),
// tileLen   = elements to move into LDS.
__device__ inline void tdm_load_1d_dwords(unsigned ldsOff, const void* gptr,
                                          unsigned tensorLen, unsigned tileLen) {
  typedef unsigned u32;
  typedef u32 v4u __attribute__((ext_vector_type(4)));
  typedef u32 v8u __attribute__((ext_vector_type(8)));
  const unsigned long long ga = (unsigned long long)(uintptr_t)gptr;

  v4u g0;
  g0.x = 1u;                                              // count=1, user mode
  g0.y = ldsOff;                                          // lds_addr
  g0.z = (u32)ga;                                         // global_addr[31:0]
  g0.w = (u32)((ga >> 32) & 0x1FFFFFFull) | (2u << 30);   // ga[56:32] | type=2

  v8u g1;
  g1.s0 = 2u << 16;                       // workgroup_mask=0, data_size=4B
  g1.s1 = (tensorLen & 0xFFFFu) << 16;    // tensor_dim0[15:0]
  g1.s2 = (tensorLen >> 16) | (1u << 16); // tensor_dim0[31:16] | tensor_dim1=1
  g1.s3 = (tileLen & 0xFFFFu) << 16;      // tile_dim0
  g1.s4 = 1u;                             // tile_dim1=1, tile_dim2 unused
  g1.s5 = tensorLen;                      // tensor_dim0_stride[31:0]
  g1.s6 = 0u;
  g1.s7 = 0u;

  asm volatile("tensor_load_to_lds %0, %1" : : "s"(g0), "s"(g1) : "memory");
}

// ---------------------------------------------------------------------------
// SpMM variant A: out[rows[e],:] += vals[e] * x[cols[e],:], COO streams
// double-buffered through LDS with per-lane async DMA loads (ASYNCcnt).
// 16 lanes per nonzero (float4 each -> one 64-float row).
// ---------------------------------------------------------------------------
#define SPMM_BLOCK 256
#define SPMM_CHUNK 256

__global__ __launch_bounds__(SPMM_BLOCK) void spmm_async_kernel(
    const int* __restrict__ rows, const int* __restrict__ cols,
    const float* __restrict__ vals, const float* __restrict__ x,
    float* __restrict__ out, int nnz)
{
  __shared__ int   s_rows[2][SPMM_CHUNK];
  __shared__ int   s_cols[2][SPMM_CHUNK];
  __shared__ float s_vals[2][SPMM_CHUNK];

  const int tid     = threadIdx.x;
  const int nChunks = (nnz + SPMM_CHUNK - 1) / SPMM_CHUNK;

  auto stage = [&](int c, int b) {
    int i = c * SPMM_CHUNK + tid;
    if (i < nnz) {
      async_load_b32_to_lds(lds_addr32(&s_rows[b][tid]), rows + i);
      async_load_b32_to_lds(lds_addr32(&s_cols[b][tid]), cols + i);
      async_load_b32_to_lds(lds_addr32(&s_vals[b][tid]), vals + i);
    }
  };

  int c = blockIdx.x;
  if (c < nChunks) stage(c, 0);

  const int grp  = tid >> 4;
  const int lane = tid & 15;
  const float4* __restrict__ x4 = (const float4*)x;
  int buf = 0;

  for (; c < nChunks; c += gridDim.x) {
    wait_asynccnt0();
    __syncthreads();
    const int nbuf = buf ^ 1;
    const int cn   = c + gridDim.x;
    if (cn < nChunks) stage(cn, nbuf);

    int limit = nnz - c * SPMM_CHUNK;
    if (limit > SPMM_CHUNK) limit = SPMM_CHUNK;

    for (int j = grp; j < limit; j += 16) {
      const int   r   = s_rows[buf][j];
      const int   col = s_cols[buf][j];
      const float a   = s_vals[buf][j];
      const float4 xv = x4[(size_t)col * 16 + lane];
      float* o = out + (size_t)r * (size_t)D_N + lane * 4;
      atomic_add_f32_dev(o + 0, a * xv.x);
      atomic_add_f32_dev(o + 1, a * xv.y);
      atomic_add_f32_dev(o + 2, a * xv.z);
      atomic_add_f32_dev(o + 3, a * xv.w);
    }
    buf = nbuf;
  }
}

// ---------------------------------------------------------------------------
// SpMM variant B: identical math; COO streams staged by the Tensor Data
// Mover (one wave issues 3 descriptors per chunk, TENSORcnt-synchronized),
// double-buffered.
// ---------------------------------------------------------------------------
#define TDM_CHUNK 1024

__global__ __launch_bounds__(SPMM_BLOCK) void spmm_tdm_kernel(
    const int* __restrict__ rows, const int* __restrict__ cols,
    const float* __restrict__ vals, const float* __restrict__ x,
    float* __restrict__ out, int nnz)
{
  __shared__ int   s_rows[2][TDM_CHUNK];
  __shared__ int   s_cols[2][TDM_CHUNK];
  __shared__ float s_vals[2][TDM_CHUNK];

  const int tid     = threadIdx.x;
  const int nChunks = (nnz + TDM_CHUNK - 1) / TDM_CHUNK;

  auto stage = [&](int c, int b) {
    // One wave issues the DMA; EXEC is ignored by tensor ops, branch keeps
    // the instruction out of the other waves' streams.
    if (tid < 32) {
      const int base = c * TDM_CHUNK;
      const unsigned remaining = (unsigned)(nnz - base);
      tdm_load_1d_dwords(lds_addr32(&s_rows[b][0]), rows + base, remaining, TDM_CHUNK);
      tdm_load_1d_dwords(lds_addr32(&s_cols[b][0]), cols + base, remaining, TDM_CHUNK);
      tdm_load_1d_dwords(lds_addr32(&s_vals[b][0]), vals + base, remaining, TDM_CHUNK);
    }
  };

  int c = blockIdx.x;
  if (c < nChunks) stage(c, 0);

  const int grp  = tid >> 4;
  const int lane = tid & 15;
  const float4* __restrict__ x4 = (const float4*)x;
  int buf = 0;

  for (; c < nChunks; c += gridDim.x) {
    if (tid < 32) __builtin_amdgcn_s_wait_tensorcnt(0);  // wave 0's DMAs done
    __syncthreads();                                     // publish to block
    const int nbuf = buf ^ 1;
    const int cn   = c + gridDim.x;
    if (cn < nChunks) stage(cn, nbuf);

    int limit = nnz - c * TDM_CHUNK;
    if (limit > TDM_CHUNK) limit = TDM_CHUNK;

    for (int j = grp; j < limit; j += 16) {
      const int   r   = s_rows[buf][j];
      const int   col = s_cols[buf][j];
      const float a   = s_vals[buf][j];
      const float4 xv = x4[(size_t)col * 16 + lane];
      float* o = out + (size_t)r * (size_t)D_N + lane * 4;
      atomic_add_f32_dev(o + 0, a * xv.x);
      atomic_add_f32_dev(o + 1, a * xv.y);
      atomic_add_f32_dev(o + 2, a * xv.z);
      atomic_add_f32_dev(o + 3, a * xv.w);
    }
    buf = nbuf;
  }
}

// ---------------------------------------------------------------------------
// Fused: x_next = dropout(prop + x_cur), acc += x_next (prop may alias xout).
// Threefry "partitionable" bits: per element i, counter=(0,i), low 32 bits.
// ---------------------------------------------------------------------------
__global__ __launch_bounds__(256) void fuse_dropout_kernel(
    const float* prop, const float* xin, float* xout, float* acc,
    unsigned fk0, unsigned fk1, int total)
{
  int i = blockIdx.x * blockDim.x + threadIdx.x;
  if (i >= total) return;

  float v = prop[i] + xin[i];

  unsigned y0, y1;
  threefry2x32_block(fk0, fk1, 0u, (unsigned)i, y0, y1);
  const unsigned bits = y1;
  const float u = __uint_as_float((bits >> 9) | 0x3f800000u) - 1.0f;
  const float r = (u < 0.5f) ? v * 2.0f : 0.0f;   // keep=0.5 -> x/keep = 2x

  xout[i] = r;
  acc[i] += r;
}

__global__ __launch_bounds__(256) void init_kernel(
    const float* __restrict__ e, float* __restrict__ x,
    float* __restrict__ acc, int total)
{
  int i = blockIdx.x * blockDim.x + threadIdx.x;
  if (i >= total) return;
  float v = e[i];
  x[i]   = v;
  acc[i] = v;
}

__global__ __launch_bounds__(256) void scale_kernel(float* a, float s, int total)
{
  int i = blockIdx.x * blockDim.x + threadIdx.x;
  if (i < total) a[i] *= s;
}

// ---------------------------------------------------------------------------
// Host orchestration
// ---------------------------------------------------------------------------
extern "C" void kernel_launch(void* const* d_in, const int* in_sizes, int n_in,
                              void* d_out, int out_size, void* d_ws, size_t ws_size,
                              hipStream_t stream) {
  (void)in_sizes; (void)n_in; (void)out_size; (void)ws_size;

  const float* embs    = (const float*)d_in[0];
  const int*   up_rows = (const int*)  d_in[1];
  const int*   up_cols = (const int*)  d_in[2];
  const float* up_vals = (const float*)d_in[3];
  const int*   pu_rows = (const int*)  d_in[4];
  const int*   pu_cols = (const int*)  d_in[5];
  const float* pu_vals = (const float*)d_in[6];

  float* acc  = (float*)d_out;                    // running sum, scaled at end
  float* bufA = (float*)d_ws;                     // [P, 64]
  float* bufB = bufA + (size_t)P_N * D_N;         // [P, 64]
  float* msg  = bufB + (size_t)P_N * D_N;         // [U, 64]

  const int total    = P_N * D_N;   // 12,800,000
  const int msgTotal = U_N * D_N;   //  3,200,000

  const int eltBlocks   = (total + 255) / 256;
  const int asyncChunks = (NNZ_N + SPMM_CHUNK - 1) / SPMM_CHUNK;
  const int asyncBlocks = asyncChunks < 2048 ? asyncChunks : 2048;
  const int tdmChunks   = (NNZ_N + TDM_CHUNK - 1) / TDM_CHUNK;
  const int tdmBlocks   = tdmChunks < 1024 ? tdmChunks : 1024;

  init_kernel<<<eltBlocks, 256, 0, stream>>>(embs, bufA, acc, total);

  float* xcur  = bufA;
  float* xnext = bufB;
  for (int layer = 0; layer < NUM_LAYERS; ++layer) {
    // folded key = threefry_block(key(42)=(0,42), counter=(0, layer))
    unsigned fk0, fk1;
    threefry2x32_block(0u, 42u, 0u, (unsigned)layer, fk0, fk1);

    hipMemsetAsync(msg, 0, (size_t)msgTotal * sizeof(float), stream);
    spmm_async_kernel<<<asyncBlocks, SPMM_BLOCK, 0, stream>>>(
        up_rows, up_cols, up_vals, xcur, msg, NNZ_N);           // [U,64]

    hipMemsetAsync(xnext, 0, (size_t)total * sizeof(float), stream);
    spmm_tdm_kernel<<<tdmBlocks, SPMM_BLOCK, 0, stream>>>(
        pu_rows, pu_cols, pu_vals, msg, xnext, NNZ_N);          // [P,64]

    fuse_dropout_kernel<<<eltBlocks, 256, 0, stream>>>(
        xnext, xcur, xnext, acc, fk0, fk1, total);

    float* t = xcur; xcur = xnext; xnext = t;
  }

  scale_kernel<<<eltBlocks, 256, 0, stream>>>(acc, 1.0f / (NUM_LAYERS + 1), total);
}